// RGCN_68092411510976
// MI455X (gfx1250) — compile-verified
//
#include <hip/hip_runtime.h>

#define DIM_IN 256
#define DIM_HID 256
#define DIM_OUT 128
#define NREL 4

typedef __attribute__((ext_vector_type(2))) float v2f;
typedef __attribute__((ext_vector_type(8))) float v8f;

// ---------------- utility kernels ----------------

__global__ void k_zero(float* __restrict__ p, int n) {
  int i = blockIdx.x * blockDim.x + threadIdx.x;
  if (i < n) p[i] = 0.0f;
}

__global__ void k_count_deg(const int* __restrict__ src, const int* __restrict__ dst,
                            float* __restrict__ outdeg, float* __restrict__ indeg,
                            int n, int e) {
  int i = blockIdx.x * blockDim.x + threadIdx.x;
  if (i >= NREL * e) return;
  int r = i / e;
  unsafeAtomicAdd(&outdeg[(size_t)r * n + src[i]], 1.0f);
  unsafeAtomicAdd(&indeg[(size_t)r * n + dst[i]], 1.0f);
}

__global__ void k_deg_to_norm(float* __restrict__ p, int n) {
  int i = blockIdx.x * blockDim.x + threadIdx.x;
  if (i < n) {
    float d = p[i];
    d = d < 1.0f ? 1.0f : d;
    p[i] = rsqrtf(d);
  }
}

// out[n][D] = sum_r bias[r][D]  (accumulator seed: biases summed across relations)
template <int D>
__global__ void k_init_bias(float* __restrict__ out, const float* __restrict__ b, int n) {
  long long i = (long long)blockIdx.x * blockDim.x + threadIdx.x;
  if (i >= (long long)n * D) return;
  int d = (int)(i % D);
  float s = 0.0f;
#pragma unroll
  for (int r = 0; r < NREL; ++r) s += b[r * D + d];
  out[i] = s;
}

// ---------------- W repack into WMMA B-fragment layout ----------------
// Bp[((r*CT + ct)*(DIN/4) + k0/4)*32 + lane] = {W[r][k0+koff][ct*16+mrow],
//                                               W[r][k0+koff+1][ct*16+mrow]}
// with mrow = lane&15, koff = (lane>>4)*2 — matches V_WMMA_F32_16X16X4_F32 B layout,
// so the hot loop issues one coalesced global_load_b64 per fragment.
template <int DIN, int DOUT>
__global__ void k_pack_w(const float* __restrict__ W, float* __restrict__ Bp) {
  constexpr int CT = DOUT / 16;
  const int total = NREL * CT * (DIN / 4) * 32;
  int i = blockIdx.x * blockDim.x + threadIdx.x;
  if (i >= total) return;
  int lane = i & 31;
  int grp = i >> 5;
  int kq = grp % (DIN / 4);
  int ct = (grp / (DIN / 4)) % CT;
  int r = grp / (DIN / 4) / CT;
  int mrow = lane & 15;
  int koff = (lane >> 4) * 2;
  int k = kq * 4 + koff;
  int col = ct * 16 + mrow;
  const float* Wr = W + (size_t)r * DIN * DOUT;
  v2f v;
  v.x = Wr[(size_t)k * DOUT + col];
  v.y = Wr[(size_t)(k + 1) * DOUT + col];
  *(v2f*)(Bp + (size_t)i * 2) = v;
}

// ---------------- WMMA GEMM: Y[r] = (op(X) @ W[r]) * outnorm[r][row] ----------------
// op = ReLU for layer 2 (fused on the LDS staging load).
// One 256-thread block (8 wave32) computes a 32-row x DOUT strip for one relation:
// MT=2 row-tiles per wave (B fragment reused across both), TPW col-tiles per wave.
template <int DIN, int DOUT, bool RELU>
__launch_bounds__(256)
__global__ void k_gemm_xw(const float* __restrict__ X,        // [n, DIN]
                          const float* __restrict__ Bp,       // packed W fragments
                          const float* __restrict__ outnorm,  // [NREL, n]
                          float* __restrict__ Y,              // [NREL, n, DOUT]
                          int n) {
  constexpr int CT = DOUT / 16;   // col tiles total (16 or 8)
  constexpr int TPW = CT / 8;     // col tiles per wave (2 or 1)
  constexpr int MT = 2;           // row tiles per wave (32 rows per block)
  __shared__ float xs[MT * 16 * DIN];  // 32 KB

  const int r = blockIdx.y;
  const int row0 = blockIdx.x * (MT * 16);
  const int tid = threadIdx.x;
  const int wave = tid >> 5;
  const int lane = tid & 31;

  // Stage 32 x DIN tile of X into LDS (ReLU fused for layer 2)
  for (int i = tid; i < MT * 16 * DIN; i += 256) {
    int rr = i / DIN, cc = i - rr * DIN;
    int row = row0 + rr;
    float v = (row < n) ? X[(size_t)row * DIN + cc] : 0.0f;
    if (RELU) v = fmaxf(v, 0.0f);
    xs[i] = v;
  }
  __syncthreads();

  const int mrow = lane & 15;        // M (A) / N (B) position within tile
  const int koff = (lane >> 4) * 2;  // lanes>=16 hold K=2,3 of each K-quad
  const int half = lane >> 4;        // C/D: lanes>=16 hold M=i+8

  v8f acc[MT][TPW];
#pragma unroll
  for (int m = 0; m < MT; ++m)
#pragma unroll
    for (int t = 0; t < TPW; ++t) acc[m][t] = {};

  // per-wave packed-B base pointers: linear, 256 B stride per K-step
  const float* bptr[TPW];
#pragma unroll
  for (int t = 0; t < TPW; ++t)
    bptr[t] = Bp + (((size_t)r * CT + wave * TPW + t) * (DIN / 4)) * 64 + lane * 2;

#pragma unroll 4
  for (int k0 = 0; k0 < DIN; k0 += 4) {
    v2f a0 = *(const v2f*)&xs[(0 * 16 + mrow) * DIN + k0 + koff];
    v2f a1 = *(const v2f*)&xs[(1 * 16 + mrow) * DIN + k0 + koff];
#pragma unroll
    for (int t = 0; t < TPW; ++t) {
      v2f b = *(const v2f*)(bptr[t] + (k0 >> 2) * 64);
      acc[0][t] = __builtin_amdgcn_wmma_f32_16x16x4_f32(
          false, a0, false, b, (short)0, acc[0][t], false, false);
      acc[1][t] = __builtin_amdgcn_wmma_f32_16x16x4_f32(
          false, a1, false, b, (short)0, acc[1][t], false, false);
    }
  }

  // Epilogue: scale row by out-degree norm (folds D_src^-1/2) and store.
  const int colbase = wave * TPW * 16;
  if (row0 + MT * 16 <= n) {  // full tile: branchless
#pragma unroll
    for (int m = 0; m < MT; ++m)
#pragma unroll
      for (int i = 0; i < 8; ++i) {
        int row = row0 + m * 16 + i + half * 8;
        float sc = outnorm[(size_t)r * n + row];
#pragma unroll
        for (int t = 0; t < TPW; ++t)
          Y[((size_t)r * n + row) * DOUT + colbase + t * 16 + mrow] = acc[m][t][i] * sc;
      }
  } else {  // ragged tail
#pragma unroll
    for (int m = 0; m < MT; ++m)
#pragma unroll
      for (int i = 0; i < 8; ++i) {
        int row = row0 + m * 16 + i + half * 8;
        if (row >= n) continue;
        float sc = outnorm[(size_t)r * n + row];
#pragma unroll
        for (int t = 0; t < TPW; ++t)
          Y[((size_t)r * n + row) * DOUT + colbase + t * 16 + mrow] = acc[m][t][i] * sc;
      }
  }
}

// ---------------- edge scatter: H[dst] += innorm[r][dst] * XWS[r][src] ----------------
// One wave32 per edge; D/32 floats per lane via float4 loads + hw fp32 atomics.
template <int D>
__launch_bounds__(256)
__global__ void k_scatter(const float* __restrict__ XWS,     // [NREL, n, D] (src-norm folded)
                          const int* __restrict__ src,       // [NREL, e] flat
                          const int* __restrict__ dst,       // [NREL, e] flat
                          const float* __restrict__ innorm,  // [NREL, n]
                          float* __restrict__ H,             // [n, D] accumulator
                          int n, int e) {
  const int lane = threadIdx.x & 31;
  const int wave = threadIdx.x >> 5;
  const long long idx = (long long)blockIdx.x * 8 + wave;
  if (idx >= (long long)NREL * e) return;
  const int r = (int)(idx / e);
  const int s = src[idx];
  const int d = dst[idx];
  const float coef = innorm[(size_t)r * n + d];
  const float* srcRow = XWS + ((size_t)r * n + s) * D;
  float* dstRow = H + (size_t)d * D;
  constexpr int PER = D / 32;  // 8 (D=256) or 4 (D=128) floats per lane
#pragma unroll
  for (int q = 0; q < PER / 4; ++q) {
    const float4 v = *(const float4*)(srcRow + lane * PER + q * 4);
    float* p = dstRow + lane * PER + q * 4;
    unsafeAtomicAdd(p + 0, v.x * coef);
    unsafeAtomicAdd(p + 1, v.y * coef);
    unsafeAtomicAdd(p + 2, v.z * coef);
    unsafeAtomicAdd(p + 3, v.w * coef);
  }
}

// ---------------- launch ----------------

extern "C" void kernel_launch(void* const* d_in, const int* in_sizes, int n_in,
                              void* d_out, int out_size, void* d_ws, size_t ws_size,
                              hipStream_t stream) {
  const float* x  = (const float*)d_in[0];
  const int* src  = (const int*)d_in[1];
  const int* dst  = (const int*)d_in[2];
  const float* W1 = (const float*)d_in[3];
  const float* b1 = (const float*)d_in[4];
  const float* W2 = (const float*)d_in[5];
  const float* b2 = (const float*)d_in[6];

  const int n = in_sizes[0] / DIM_IN;   // 100000
  const int e = in_sizes[1] / NREL;     // 400000

  // workspace (floats): [outnorm R*n][innorm R*n][Bp R*256*256][xws R*n*256][H n*256]
  float* outn = (float*)d_ws;
  float* inn  = outn + (size_t)NREL * n;
  float* Bp   = inn + (size_t)NREL * n;
  float* xws  = Bp + (size_t)NREL * DIM_IN * DIM_HID;
  float* H    = xws + (size_t)NREL * n * DIM_HID;
  float* out  = (float*)d_out;

  const int re = NREL * e;
  const int degElems = 2 * NREL * n;
  const int rowTiles = (n + 31) / 32;

  // degree norms (shared by both layers)
  k_zero<<<(degElems + 255) / 256, 256, 0, stream>>>(outn, degElems);
  k_count_deg<<<(re + 255) / 256, 256, 0, stream>>>(src, dst, outn, inn, n, e);
  k_deg_to_norm<<<(degElems + 255) / 256, 256, 0, stream>>>(outn, degElems);

  // ---- layer 1: H = sum_r b1_r, then scatter norm'd X@W1_r messages ----
  {
    const int packElems = NREL * (DIM_HID / 16) * (DIM_IN / 4) * 32;
    k_pack_w<DIM_IN, DIM_HID><<<(packElems + 255) / 256, 256, 0, stream>>>(W1, Bp);
  }
  k_init_bias<DIM_HID><<<(unsigned)(((long long)n * DIM_HID + 255) / 256), 256, 0, stream>>>(H, b1, n);
  k_gemm_xw<DIM_IN, DIM_HID, false><<<dim3(rowTiles, NREL), 256, 0, stream>>>(x, Bp, outn, xws, n);
  k_scatter<DIM_HID><<<(re + 7) / 8, 256, 0, stream>>>(xws, src, dst, inn, H, n, e);

  // ---- layer 2: out = sum_r b2_r + scatter of relu(H)@W2_r (ReLU fused in GEMM) ----
  {
    const int packElems = NREL * (DIM_OUT / 16) * (DIM_HID / 4) * 32;
    k_pack_w<DIM_HID, DIM_OUT><<<(packElems + 255) / 256, 256, 0, stream>>>(W2, Bp);
  }
  k_init_bias<DIM_OUT><<<(unsigned)(((long long)n * DIM_OUT + 255) / 256), 256, 0, stream>>>(out, b2, n);
  k_gemm_xw<DIM_HID, DIM_OUT, true><<<dim3(rowTiles, NREL), 256, 0, stream>>>(H, Bp, outn, xws, n);
  k_scatter<DIM_OUT><<<(re + 7) / 8, 256, 0, stream>>>(xws, src, dst, inn, out, n, e);
}